// GPT2Attention_3951369912913
// MI455X (gfx1250) — compile-verified
//
#include <hip/hip_runtime.h>
#include <hip/hip_bf16.h>
#include <math.h>
#include <stdint.h>

// ---------------------------------------------------------------------------
// GPT-2 attention, fp32 end-to-end, CDNA5 V_WMMA_F32_16X16X4_F32 +
// GLOBAL_LOAD_ASYNC_TO_LDS_B128 double-buffered staging (ASYNCcnt).
//   B=2, S=2048, HIDDEN=2048, NUM_HEADS=16, HEAD=128
// ---------------------------------------------------------------------------

#define BATCH   2
#define SEQ     2048
#define HIDDEN  2048
#define NH      16
#define HEAD    128
#define N3      (3 * HIDDEN)          // 6144
#define M_TOT   (BATCH * SEQ)         // 4096

typedef float v2f __attribute__((ext_vector_type(2)));
typedef float v8f __attribute__((ext_vector_type(8)));
typedef int   v4i __attribute__((vector_size(16)));

// ---- async copy to LDS (CDNA5) with synchronous fallback ------------------
#if defined(__gfx1250__) && __has_builtin(__builtin_amdgcn_global_load_async_to_lds_b128) && \
    __has_builtin(__builtin_amdgcn_s_wait_asynccnt)
#define USE_ASYNC 1
#else
#define USE_ASYNC 0
#endif

typedef __attribute__((address_space(1))) v4i GASv4i;   // global (prints as __device__)
typedef __attribute__((address_space(3))) v4i LASv4i;   // LDS    (prints as __shared__)

__device__ __forceinline__ void copy16(float* lds, const float* g) {
#if USE_ASYNC
    // AS1 shares the flat VA for global memory -> inttoptr is safe for g;
    // LDS side is a real addrspacecast (generic -> local).
    __builtin_amdgcn_global_load_async_to_lds_b128(
        (GASv4i*)(uintptr_t)g, (LASv4i*)lds, 0, 0);
#else
    *(float4*)lds = *(const float4*)g;
#endif
}

__device__ __forceinline__ void async_join() {
#if USE_ASYNC
    __builtin_amdgcn_s_wait_asynccnt(0);
#endif
}

// ===========================================================================
// Kernel 1: QKV projection GEMM.  M=4096, N=6144, K=2048.
// Block tile 128x128, 8 waves; wave tile 64x32 -> 4x2 wmma accumulators.
// Double-buffered LDS, async staging.
// ===========================================================================
#define BM 128
#define BN 128
#define BK 32
#define LDA 36     // As row stride in floats (16B aligned, bank-friendly)
#define LDB 132    // Bs row stride in floats

__global__ __launch_bounds__(256)
void qkv_gemm_kernel(const float* __restrict__ enc,
                     const float* __restrict__ w,
                     const float* __restrict__ bias,
                     float* __restrict__ q_ws,
                     float* __restrict__ k_ws,
                     float* __restrict__ v_ws)
{
    __shared__ float As[2][BM * LDA];   // 2 x 18.4 KB
    __shared__ float Bs[2][BK * LDB];   // 2 x 16.9 KB

    const int tid  = threadIdx.x;
    const int lane = tid & 31;
    const int wave = tid >> 5;       // 0..7
    const int wm   = wave >> 2;      // 0..1 : 64-row strip
    const int wn   = wave & 3;       // 0..3 : 32-col strip
    const int half = lane >> 4;      // 0/1 (k-group within wmma operand)
    const int l15  = lane & 15;

    const int bm0 = blockIdx.y * BM;
    const int bn0 = blockIdx.x * BN;

    v8f acc[4][2];
#pragma unroll
    for (int mt = 0; mt < 4; ++mt)
#pragma unroll
        for (int nt = 0; nt < 2; ++nt)
            acc[mt][nt] = (v8f){0.f, 0.f, 0.f, 0.f, 0.f, 0.f, 0.f, 0.f};

    // --- staging helper: issue 8 x 16B copies for slab k0 into buf
    auto stage = [&](int buf, int k0) {
#pragma unroll
        for (int i = 0; i < 4; ++i) {
            int idx = tid + i * 256;
            int r   = idx >> 3;          // /8 float4 per A row
            int c4  = idx & 7;
            copy16(&As[buf][r * LDA + c4 * 4],
                   enc + (size_t)(bm0 + r) * HIDDEN + k0 + c4 * 4);
        }
#pragma unroll
        for (int i = 0; i < 4; ++i) {
            int idx = tid + i * 256;
            int r   = idx >> 5;          // /32 float4 per B row
            int c4  = idx & 31;
            copy16(&Bs[buf][r * LDB + c4 * 4],
                   w + (size_t)(k0 + r) * N3 + bn0 + c4 * 4);
        }
    };

    int cur = 0;
    stage(cur, 0);
    async_join();
    __syncthreads();

    for (int k0 = 0; k0 < HIDDEN; k0 += BK) {
        if (k0 + BK < HIDDEN) stage(cur ^ 1, k0 + BK);

        const float* Ab = As[cur];
        const float* Bb = Bs[cur];
#pragma unroll
        for (int kk = 0; kk < BK; kk += 4) {
            const int ks = kk + half * 2;
            v2f afrag[4];
#pragma unroll
            for (int mt = 0; mt < 4; ++mt) {
                int m = wm * 64 + mt * 16 + l15;
                afrag[mt].x = Ab[m * LDA + ks];
                afrag[mt].y = Ab[m * LDA + ks + 1];
            }
            v2f bfrag[2];
#pragma unroll
            for (int nt = 0; nt < 2; ++nt) {
                int n = wn * 32 + nt * 16 + l15;
                bfrag[nt].x = Bb[ks * LDB + n];
                bfrag[nt].y = Bb[(ks + 1) * LDB + n];
            }
#pragma unroll
            for (int mt = 0; mt < 4; ++mt)
#pragma unroll
                for (int nt = 0; nt < 2; ++nt)
                    acc[mt][nt] = __builtin_amdgcn_wmma_f32_16x16x4_f32(
                        false, afrag[mt], false, bfrag[nt],
                        (short)0, acc[mt][nt], false, false);
        }

        async_join();       // next slab resident in LDS
        __syncthreads();    // all waves done reading cur / writing nxt
        cur ^= 1;
    }

    // Epilogue: add bias, scatter into head-major Q/K/V workspace.
    // Reference splits qkv as [key | query | value] along columns.
#pragma unroll
    for (int mt = 0; mt < 4; ++mt) {
#pragma unroll
        for (int nt = 0; nt < 2; ++nt) {
            const int n  = bn0 + wn * 32 + nt * 16 + l15;
            const float bb = bias[n];
            const int sec = n >> 11;              // /2048: 0=K,1=Q,2=V
            const int hid = n & (HIDDEN - 1);
            const int hh  = hid >> 7;             // head
            const int ch  = hid & (HEAD - 1);     // channel
            float* base = (sec == 0) ? k_ws : (sec == 1) ? q_ws : v_ws;
#pragma unroll
            for (int j = 0; j < 8; ++j) {
                const int m = bm0 + wm * 64 + mt * 16 + j + half * 8;
                const int b = m >> 11;            // /2048
                const int s = m & (SEQ - 1);
                base[(((size_t)b * NH + hh) * SEQ + s) * HEAD + ch] = acc[mt][nt][j] + bb;
            }
        }
    }
}

// ===========================================================================
// Kernel 2: flash attention. Block = 4 waves = 64 query rows of one (b,h).
// Double-buffered K/V tiles, async staging.
// ===========================================================================
#define QT 64      // query rows per block
#define KT 64      // key tile
#define LQ 132
#define LK 132
#define LV 132
#define LP 68

__global__ __launch_bounds__(128)
void flash_attn_kernel(const float* __restrict__ q_ws,
                       const float* __restrict__ k_ws,
                       const float* __restrict__ v_ws,
                       const float* __restrict__ mask,
                       float* __restrict__ out)
{
    __shared__ float Qs[QT * LQ];          // 33.8 KB
    __shared__ float Ks[2][KT * LK];       // 67.6 KB
    __shared__ float Vs[2][KT * LV];       // 67.6 KB
    __shared__ float Ps[4 * 16 * LP];      // 17.4 KB  (per-wave P scratch)

    const int tid  = threadIdx.x;
    const int lane = tid & 31;
    const int wave = tid >> 5;             // 0..3
    const int half = lane >> 4;
    const int l15  = lane & 15;

    const int bh = blockIdx.y;             // b*NH + h
    const int q0 = blockIdx.x * QT;

    const float* Qg = q_ws + (size_t)bh * SEQ * HEAD;
    const float* Kg = k_ws + (size_t)bh * SEQ * HEAD;
    const float* Vg = v_ws + (size_t)bh * SEQ * HEAD;

    auto stage_kv = [&](int buf, int kt0) {
#pragma unroll
        for (int i = 0; i < 16; ++i) {
            int idx = tid + i * 128;
            int r   = idx >> 5;
            int c4  = idx & 31;
            copy16(&Ks[buf][r * LK + c4 * 4],
                   Kg + (size_t)(kt0 + r) * HEAD + c4 * 4);
            copy16(&Vs[buf][r * LV + c4 * 4],
                   Vg + (size_t)(kt0 + r) * HEAD + c4 * 4);
        }
    };

    // Stage Q tile (once) + first K/V tile.
#pragma unroll
    for (int i = 0; i < 16; ++i) {
        int idx = tid + i * 128;
        int r   = idx >> 5;
        int c4  = idx & 31;
        copy16(&Qs[r * LQ + c4 * 4], Qg + (size_t)(q0 + r) * HEAD + c4 * 4);
    }
    int cur = 0;
    stage_kv(cur, 0);
    async_join();
    __syncthreads();

    const float scale = 0.08838834764831845f;   // 1/sqrt(128)

    float m_run[8], l_run[8];
#pragma unroll
    for (int j = 0; j < 8; ++j) { m_run[j] = -1e30f; l_run[j] = 0.f; }
    v8f o[8];
#pragma unroll
    for (int ot = 0; ot < 8; ++ot)
        o[ot] = (v8f){0.f, 0.f, 0.f, 0.f, 0.f, 0.f, 0.f, 0.f};

    float* Pw = Ps + wave * 16 * LP;

    for (int kt0 = 0; kt0 < SEQ; kt0 += KT) {
        if (kt0 + KT < SEQ) stage_kv(cur ^ 1, kt0 + KT);
        const float* Kb = Ks[cur];
        const float* Vb = Vs[cur];

        // --- scores: S = Q(16x128) * K_tile^T(128x64), 4 wmma column tiles ---
        v8f st[4];
#pragma unroll
        for (int nt = 0; nt < 4; ++nt)
            st[nt] = (v8f){0.f, 0.f, 0.f, 0.f, 0.f, 0.f, 0.f, 0.f};

        for (int kk = 0; kk < HEAD; kk += 4) {
            const int ks = kk + half * 2;
            v2f aq;
            const int m = wave * 16 + l15;
            aq.x = Qs[m * LQ + ks];
            aq.y = Qs[m * LQ + ks + 1];
#pragma unroll
            for (int nt = 0; nt < 4; ++nt) {
                v2f bk;                         // B[k][n] = K_tile[key n][k]
                const int key = nt * 16 + l15;
                bk.x = Kb[key * LK + ks];
                bk.y = Kb[key * LK + ks + 1];
                st[nt] = __builtin_amdgcn_wmma_f32_16x16x4_f32(
                    false, aq, false, bk, (short)0, st[nt], false, false);
            }
        }

        // --- scale + multiplicative mask (faithful to reference) ---
#pragma unroll
        for (int nt = 0; nt < 4; ++nt) {
#pragma unroll
            for (int j = 0; j < 8; ++j) {
                const int qg = q0 + wave * 16 + j + half * 8;
                const int kg = kt0 + nt * 16 + l15;
                st[nt][j] = st[nt][j] * scale * mask[(size_t)qg * SEQ + kg];
            }
        }

        // --- online softmax (rows live as: VGPR j, lane-half -> row j+8*half) ---
#pragma unroll
        for (int j = 0; j < 8; ++j) {
            float mx = fmaxf(fmaxf(st[0][j], st[1][j]), fmaxf(st[2][j], st[3][j]));
            mx = fmaxf(mx, __shfl_xor(mx, 1, 32));
            mx = fmaxf(mx, __shfl_xor(mx, 2, 32));
            mx = fmaxf(mx, __shfl_xor(mx, 4, 32));
            mx = fmaxf(mx, __shfl_xor(mx, 8, 32));
            const float m_new = fmaxf(m_run[j], mx);
            const float alpha = __expf(m_run[j] - m_new);
            float rs = 0.f;
#pragma unroll
            for (int nt = 0; nt < 4; ++nt) {
                const float p = __expf(st[nt][j] - m_new);
                st[nt][j] = p;
                rs += p;
            }
            rs += __shfl_xor(rs, 1, 32);
            rs += __shfl_xor(rs, 2, 32);
            rs += __shfl_xor(rs, 4, 32);
            rs += __shfl_xor(rs, 8, 32);
            l_run[j] = l_run[j] * alpha + rs;
            m_run[j] = m_new;
#pragma unroll
            for (int ot = 0; ot < 8; ++ot) o[ot][j] *= alpha;
        }

        // --- C-layout -> A-layout round trip through per-wave LDS ---
#pragma unroll
        for (int nt = 0; nt < 4; ++nt)
#pragma unroll
            for (int j = 0; j < 8; ++j)
                Pw[(j + half * 8) * LP + nt * 16 + l15] = st[nt][j];

        // --- O += P(16x64) * V_tile(64x128) ---
#pragma unroll
        for (int kk = 0; kk < KT; kk += 4) {
            const int ks = kk + half * 2;
            v2f ap;
            ap.x = Pw[l15 * LP + ks];
            ap.y = Pw[l15 * LP + ks + 1];
#pragma unroll
            for (int ot = 0; ot < 8; ++ot) {
                v2f bv;                         // B[k][n] = V_tile[key k][ch n]
                bv.x = Vb[ks * LV + ot * 16 + l15];
                bv.y = Vb[(ks + 1) * LV + ot * 16 + l15];
                o[ot] = __builtin_amdgcn_wmma_f32_16x16x4_f32(
                    false, ap, false, bv, (short)0, o[ot], false, false);
            }
        }

        async_join();       // next K/V tile resident
        __syncthreads();    // all waves done with cur buffers
        cur ^= 1;
    }

    // --- normalize and write out: ctx^T -> [B, S, h*128 + ch] ---
    const int b = bh >> 4;
    const int h = bh & (NH - 1);
#pragma unroll
    for (int j = 0; j < 8; ++j) {
        const float inv_l = 1.f / l_run[j];
        const int qg = q0 + wave * 16 + j + half * 8;
#pragma unroll
        for (int ot = 0; ot < 8; ++ot) {
            const int ch = ot * 16 + l15;
            out[((size_t)b * SEQ + qg) * HIDDEN + h * HEAD + ch] = o[ot][j] * inv_l;
        }
    }
}

// ===========================================================================
extern "C" void kernel_launch(void* const* d_in, const int* in_sizes, int n_in,
                              void* d_out, int out_size, void* d_ws, size_t ws_size,
                              hipStream_t stream) {
    const float* enc  = (const float*)d_in[0];   // [B,S,HIDDEN]
    const float* mask = (const float*)d_in[1];   // [1,1,S,S]
    const float* w    = (const float*)d_in[2];   // [HIDDEN, 3*HIDDEN]
    const float* bias = (const float*)d_in[3];   // [3*HIDDEN]
    float* out = (float*)d_out;                  // [B,S,HIDDEN]

    const size_t per_tensor = (size_t)BATCH * NH * SEQ * HEAD;  // 8,388,608 floats
    float* q_ws = (float*)d_ws;
    float* k_ws = q_ws + per_tensor;
    float* v_ws = k_ws + per_tensor;

    dim3 g1(N3 / BN, M_TOT / BM);    // 48 x 32
    qkv_gemm_kernel<<<g1, 256, 0, stream>>>(enc, w, bias, q_ws, k_ws, v_ws);

    dim3 g2(SEQ / QT, BATCH * NH);   // 32 x 32
    flash_attn_kernel<<<g2, 128, 0, stream>>>(q_ws, k_ws, v_ws, mask, out);
}